// HeteroFeatEncode_13769665151129
// MI455X (gfx1250) — compile-verified
//
#include <hip/hip_runtime.h>
#include <hip/hip_bf16.h>

typedef __attribute__((ext_vector_type(2))) float v2f;
typedef __attribute__((ext_vector_type(8))) float v8f;

#define NUM_TYPES 8
#define TIME_DIM  100
#define FEAT_DIM  172
#define OUT_DIM   128
#define D_IN      272            // FEAT_DIM + TIME_DIM
#define K_PAD     288            // D_IN rounded up to multiple of KC
#define TILE_M    128
#define KC        32
#define APAD      34             // A slab row stride (even -> aligned float2; conflict-free)
#define KPAIRS    (KC / 2)
#define BSTRIDE   288            // B kpair stride: 256 data + 32 pad (mod 64 == 32 -> bank-split)

// ---- CDNA5 async global->LDS path (guarded; falls back to plain stores) ----
#if defined(__has_builtin)
#if __has_builtin(__builtin_amdgcn_global_load_async_to_lds_b32) && \
    __has_builtin(__builtin_amdgcn_s_wait_asynccnt)
#define USE_ASYNC_LDS 1
#endif
#endif

typedef __attribute__((address_space(1))) int as1_i32;
typedef __attribute__((address_space(3))) int as3_i32;

__device__ __forceinline__ void lds_copy_b32(const float* g, float* l) {
#ifdef USE_ASYNC_LDS
    __builtin_amdgcn_global_load_async_to_lds_b32(
        (as1_i32*)g, (as3_i32*)l, /*offset=*/0, /*cpol=*/0);
#else
    *l = *g;
#endif
}

__device__ __forceinline__ void lds_copy_fence() {
#ifdef USE_ASYNC_LDS
    __builtin_amdgcn_s_wait_asynccnt(0);
#endif
}

// ---------------- bucketing kernels ----------------

__global__ void k_zero_counts(int* __restrict__ counts) {
    if (threadIdx.x < NUM_TYPES) counts[threadIdx.x] = 0;
}

__global__ void k_count(const int* __restrict__ types, int* __restrict__ counts, int E) {
    int i = blockIdx.x * blockDim.x + threadIdx.x;
    if (i < E) atomicAdd(&counts[types[i] & 7], 1);
}

__global__ void k_scan(const int* __restrict__ counts, int* __restrict__ seg,
                       int* __restrict__ cursor) {
    if (threadIdx.x == 0 && blockIdx.x == 0) {
        int acc = 0;
        for (int t = 0; t < NUM_TYPES; ++t) {
            seg[t] = acc;
            cursor[t] = acc;
            acc += counts[t];
        }
        seg[NUM_TYPES] = acc;
    }
}

__global__ void k_scatter(const int* __restrict__ types, int* __restrict__ cursor,
                          int* __restrict__ idx_sorted, int E) {
    int i = blockIdx.x * blockDim.x + threadIdx.x;
    if (i < E) {
        int p = atomicAdd(&cursor[types[i] & 7], 1);
        idx_sorted[p] = i;
    }
}

// ---------------- fused encode + per-type GEMM ----------------
// Block: 256 threads (8 wave32) -> one 128(M) x 128(N) output tile of one type.
// A slab (128 x 32) : async-gathered feats + on-the-fly cos time features.
// B slab (32 x 128) : W[type] in fragment-ready interleaved layout
//                     Bs2[kpair][n*2 + (k&1)] so each WMMA B operand is one
//                     aligned ds_load_b64 (no repacking movs), kpair stride 288
//                     keeps lane halves on disjoint bank sets.
// Core: V_WMMA_F32_16X16X4_F32 (exact f32 reference semantics).

__global__ __launch_bounds__(256) void k_hetero_gemm(
    const float* __restrict__ feats,      // [E,172]
    const float* __restrict__ ts,         // [E]
    const float* __restrict__ W,          // [8,272,128]
    const float* __restrict__ bias,       // [8,128]
    const float* __restrict__ emb,        // [8,128]
    const int*   __restrict__ idx_sorted, // [E] edge ids grouped by type
    const int*   __restrict__ seg,        // [9] exclusive prefix of per-type counts
    float*       __restrict__ out)        // [E,128]
{
    __shared__ float As[TILE_M * APAD];        // 17.0 KB
    __shared__ float Bs2[KPAIRS * BSTRIDE];    // 18.0 KB
    __shared__ float bwm[104];
    __shared__ float rowTs[TILE_M];
    __shared__ int   rowId[TILE_M];

    const int t   = blockIdx.y;
    const int s0  = seg[t];
    const int cnt = seg[t + 1] - s0;
    const int m0  = blockIdx.x * TILE_M;
    if (m0 >= cnt) return;

    const int tid = threadIdx.x;

    // time-encode frequencies: mult(t) * 10^(-9*j/99) = mult * exp2(-j*(9/99)*log2(10))
    if (tid < TIME_DIM) {
        const float mult = 1.0f + 0.1f * (float)t;
        bwm[tid] = mult * exp2f(-0.30199349f * (float)tid);
    }
    if (tid < TILE_M) {
        int g = m0 + tid;
        if (g < cnt) {
            int e = idx_sorted[s0 + g];
            rowId[tid] = e;
            rowTs[tid] = ts[e];
        } else {
            rowId[tid] = -1;
            rowTs[tid] = 0.0f;
        }
    }
    __syncthreads();

    const int wave = tid >> 5;            // 0..7 -> M strip [16*wave, 16*wave+16)
    const int lane = tid & 31;
    const int lm   = lane & 15;
    const int kh   = (lane >> 4) << 1;    // 0 (lanes 0-15) or 2 (lanes 16-31)
    const float* Arow = &As[(wave * 16 + lm) * APAD];

    v8f acc[8] = {};

    for (int k0 = 0; k0 < K_PAD; k0 += KC) {
        // ---- A slab: per-lane async gather of feats; cos time features via VALU ----
        #pragma unroll
        for (int i = tid; i < TILE_M * KC; i += 256) {
            int r  = i >> 5;
            int kk = i & (KC - 1);
            int k  = k0 + kk;
            int e  = rowId[r];
            float* dst = &As[r * APAD + kk];
            if (k < FEAT_DIM && e >= 0) {
                lds_copy_b32(&feats[(long)e * FEAT_DIM + k], dst);   // async DMA to LDS
            } else {
                float v = 0.0f;
                if (k < D_IN && e >= 0) v = cosf(rowTs[r] * bwm[k - FEAT_DIM]);
                *dst = v;
            }
        }
        // ---- B slab: async copy W[t] into fragment-ready interleaved layout ----
        #pragma unroll
        for (int i = tid; i < KC * OUT_DIM; i += 256) {
            int kk = i >> 7;
            int n  = i & (OUT_DIM - 1);
            int k  = k0 + kk;
            float* dst = &Bs2[(kk >> 1) * BSTRIDE + n * 2 + (kk & 1)];
            if (k < D_IN) {
                lds_copy_b32(&W[((long)t * D_IN + k) * OUT_DIM + n], dst);
            } else {
                *dst = 0.0f;
            }
        }
        lds_copy_fence();                 // s_wait_asynccnt 0 (per-wave)
        __syncthreads();

        // ---- 16x16x4 f32 WMMA over this K slab ----
        #pragma unroll
        for (int kk = 0; kk < KC; kk += 4) {
            // A frag (16x4): lanes 0-15 K=kk..kk+1, lanes 16-31 K=kk+2..kk+3
            v2f a = *(const v2f*)&Arow[kk + kh];
            const float* Bp = &Bs2[((kk + kh) >> 1) * BSTRIDE];
            #pragma unroll
            for (int nt = 0; nt < 8; ++nt) {
                v2f bb = *(const v2f*)&Bp[(nt * 16 + lm) * 2];   // single ds_load_b64
                acc[nt] = __builtin_amdgcn_wmma_f32_16x16x4_f32(
                    /*neg_a=*/false, a, /*neg_b=*/false, bb,
                    /*c_mod=*/(short)0, acc[nt],
                    /*reuse_a=*/false, /*reuse_b=*/false);
            }
        }
        __syncthreads();
    }

    // ---- epilogue: add b[t] + type_emb[t], scatter rows by edge id ----
    const int hi8 = (lane >> 4) << 3;     // C layout: VGPR r -> M=r (lo lanes) / r+8 (hi lanes)
    #pragma unroll
    for (int nt = 0; nt < 8; ++nt) {
        int n = nt * 16 + lm;
        float bn = bias[t * OUT_DIM + n] + emb[t * OUT_DIM + n];
        #pragma unroll
        for (int r = 0; r < 8; ++r) {
            int m = wave * 16 + r + hi8;
            int e = rowId[m];
            if (e >= 0) out[(long)e * OUT_DIM + n] = acc[nt][r] + bn;
        }
    }
}

// ---------------- launcher ----------------

extern "C" void kernel_launch(void* const* d_in, const int* in_sizes, int n_in,
                              void* d_out, int out_size, void* d_ws, size_t ws_size,
                              hipStream_t stream) {
    const float* feats = (const float*)d_in[0];
    const float* ts    = (const float*)d_in[1];
    const int*   types = (const int*)  d_in[2];
    const float* W     = (const float*)d_in[3];
    const float* b     = (const float*)d_in[4];
    const float* emb   = (const float*)d_in[5];
    float* out = (float*)d_out;

    const int E = in_sizes[1];            // edge_ts element count

    // workspace layout: [E] sorted ids | [8] counts | [9] seg | [8] cursor
    char* ws = (char*)d_ws;
    int* idx_sorted = (int*)ws;
    int* counts     = (int*)(ws + (size_t)E * sizeof(int));
    int* seg        = counts + NUM_TYPES;
    int* cursor     = seg + (NUM_TYPES + 1);

    const int thr = 256;
    const int blk = (E + thr - 1) / thr;

    k_zero_counts<<<1, 32, 0, stream>>>(counts);
    k_count<<<blk, thr, 0, stream>>>(types, counts, E);
    k_scan<<<1, 1, 0, stream>>>(counts, seg, cursor);
    k_scatter<<<blk, thr, 0, stream>>>(types, cursor, idx_sorted, E);

    dim3 grid((E + TILE_M - 1) / TILE_M, NUM_TYPES);
    k_hetero_gemm<<<grid, 256, 0, stream>>>(feats, ts, W, b, emb,
                                            idx_sorted, seg, out);
}